// Model_64424509440717
// MI455X (gfx1250) — compile-verified
//
#include <hip/hip_runtime.h>
#include <math.h>

// ---- problem constants ----
#define B_   64
#define L_   8192
#define C_   32
#define F_   4097
#define FP_  4160           // padded row stride for spectrum rows
#define S_   (B_*C_)        // 2048 sequences
#define N1_  64
#define N2_  128
#define KTOP 819            // int(F * 0.2)
#define TAU_ 0.5f
#define TWOPI 6.28318530717958647692f

typedef float v2f __attribute__((ext_vector_type(2)));
typedef float v8f __attribute__((ext_vector_type(8)));
typedef unsigned int u32x4 __attribute__((ext_vector_type(4)));
typedef int i32x4 __attribute__((ext_vector_type(4)));
typedef int i32x8 __attribute__((ext_vector_type(8)));

// CDNA5 f32 WMMA: D(16x16) = A(16x4) * B(4x16) + C  -- full fp32 DFT path
static __device__ __forceinline__ v8f wmma4(v2f a, v2f b, v8f c) {
  return __builtin_amdgcn_wmma_f32_16x16x4_f32(false, a, false, b, (short)0, c,
                                               false, false);
}

// ---------------------------------------------------------------------------
// Tensor Data Mover: 1D row (nelem f32) global -> LDS. One issue per wave;
// EXEC is ignored by TDM, tracked with TENSORcnt.
// D# group0: [0]=count=1, [1]=lds_addr, [2]=ga[31:0], [3]=ga[56:32]|type(2)<<30
// D# group1: data_size=4B, tensor_dim0=tile_dim0=nelem, tensor_dim1=tile_dim1=1,
//            tensor_dim0_stride=nelem
// ---------------------------------------------------------------------------
#if __has_builtin(__builtin_amdgcn_tensor_load_to_lds)
#define HAVE_TDM 1
static __device__ __forceinline__ void tdm_load_1d(const void* gptr, void* lptr,
                                                   unsigned nelem) {
  const unsigned long long ga = (unsigned long long)gptr;
  const unsigned lds = (unsigned)(unsigned long long)lptr;  // LDS aperture: low 32 bits
  u32x4 g0;
  g0[0] = 1u;                                         // count=1, user descriptor
  g0[1] = lds;                                        // lds_addr (bytes)
  g0[2] = (unsigned)(ga & 0xFFFFFFFFull);             // global_addr[31:0]
  g0[3] = (unsigned)((ga >> 32) & 0x01FFFFFFull) | (2u << 30);  // [56:32] | type=2
  i32x8 g1;
  g1[0] = 0x00020000;                                 // data_size = 2 (4 bytes)
  g1[1] = (int)((nelem & 0xFFFFu) << 16);             // tensor_dim0[15:0] @ bits 63:48
  g1[2] = (int)(((nelem >> 16) & 0xFFFFu) | (1u << 16)); // dim0[31:16] | tensor_dim1=1
  g1[3] = (int)((nelem & 0xFFFFu) << 16);             // tile_dim0 @ bits 127:112
  g1[4] = 1;                                          // tile_dim1 = 1, tile_dim2 = 0
  g1[5] = (int)nelem;                                 // tensor_dim0_stride[31:0]
  g1[6] = 0;
  g1[7] = 0;
  const i32x4 gz = {0, 0, 0, 0};
#if defined(__clang_major__) && (__clang_major__ >= 23)
  const i32x8 gz8 = {0, 0, 0, 0, 0, 0, 0, 0};
  __builtin_amdgcn_tensor_load_to_lds(g0, g1, gz, gz, gz8, 0);
#else
  __builtin_amdgcn_tensor_load_to_lds(g0, g1, gz, gz, 0);
#endif
}
#else
#define HAVE_TDM 0
#endif

// ---------------------------------------------------------------------------
// K0: x[B,L,C] -> xT[seq = b*C+c][l]   (LDS tiled transpose)
// ---------------------------------------------------------------------------
__global__ __launch_bounds__(256) void transpose_in(const float* __restrict__ x,
                                                    float* __restrict__ xT) {
  __shared__ float tile[32][33];
  const int l0 = blockIdx.x * 32;
  const int b  = blockIdx.y;
  const int r  = threadIdx.x >> 5;
  const int c  = threadIdx.x & 31;
  for (int i = r; i < 32; i += 8)
    tile[i][c] = x[((size_t)b * L_ + l0 + i) * C_ + c];   // coalesced over c
  __syncthreads();
  for (int i = r; i < 32; i += 8)                          // i = channel now
    xT[((size_t)(b * C_ + i)) * L_ + l0 + c] = tile[c][i]; // coalesced over l
}

// ---------------------------------------------------------------------------
// K1: extract diag(w1[0]), diag(w1[1])
// ---------------------------------------------------------------------------
__global__ void diag_extract(const float* __restrict__ w1,
                             float* __restrict__ diag) {
  const int i = blockIdx.x * 256 + threadIdx.x;
  if (i < 2 * F_) {
    const int j = i / F_, k = i % F_;
    diag[i] = w1[(size_t)j * F_ * F_ + (size_t)k * (F_ + 1)];
  }
}

__global__ void ent_init(float* __restrict__ e) {
  if (threadIdx.x < B_) e[threadIdx.x] = 0.0f;
}

// ---------------------------------------------------------------------------
// K2: forward four-step DFT via f32 WMMA.  n = 128*n1 + n2, k = k1 + 64*k2
//   stage1: Y[k1][n2] = sum_n1 x[n1][n2] e^{-2pi i k1 n1/64}
//   twiddle: Z = Y * e^{-2pi i k1 n2 / 8192}
//   stage2: X[k1][k2] = sum_n2 Z[k1][n2] e^{-2pi i n2 k2/128}
// keep rfft bins k <= 4096.  Input row staged into LDS by the TDM.
// ---------------------------------------------------------------------------
__global__ __launch_bounds__(256) void dft_forward(const float* __restrict__ xT,
                                                   float* __restrict__ Xr,
                                                   float* __restrict__ Xi) {
  __shared__ float s_x [N1_ * N2_];
  __shared__ float s_zr[N1_ * N2_];
  __shared__ float s_zi[N1_ * N2_];
  const int seq  = blockIdx.x;
  const int tid  = threadIdx.x;
  const int wave = tid >> 5;
  const int lane = tid & 31;
  const int lm   = lane & 15;
  const int lh   = lane >> 4;

  const float* xseq = xT + (size_t)seq * L_;
#if HAVE_TDM
  if (wave == 0) {
    tdm_load_1d(xseq, s_x, (unsigned)L_);
    __builtin_amdgcn_s_wait_tensorcnt(0);
  }
#else
  for (int n = tid; n < L_; n += 256) s_x[n] = xseq[n];
#endif
  __syncthreads();

  // ---- stage 1: 4x8 tiles of 16x16, K = 64 ----
  for (int t = wave; t < 32; t += 8) {
    const int tm = t >> 3, tn = t & 7;
    const int k1  = tm * 16 + lm;          // A row (all lanes)
    const int n2b = tn * 16 + lm;          // B col
    v8f aR = {}, aI = {};
    for (int ks = 0; ks < 16; ++ks) {
      v2f ac, as, bx;
#pragma unroll
      for (int j = 0; j < 2; ++j) {
        const int n1 = ks * 4 + lh * 2 + j;
        const int p  = (k1 * n1) & 63;
        const float th = -TWOPI * (float)p * (1.0f / 64.0f);
        ac[j] = __cosf(th);
        as[j] = __sinf(th);
        bx[j] = s_x[n1 * N2_ + n2b];
      }
      aR = wmma4(ac, bx, aR);
      aI = wmma4(as, bx, aI);
    }
#pragma unroll
    for (int v = 0; v < 8; ++v) {          // C/D layout: row = v + 8*half
      const int row = tm * 16 + v + lh * 8;
      const int col = tn * 16 + lm;
      const int p   = (row * col) & 8191;
      const float th  = -TWOPI * (float)p * (1.0f / 8192.0f);
      const float twr = __cosf(th), twi = __sinf(th);
      s_zr[row * N2_ + col] = aR[v] * twr - aI[v] * twi;
      s_zi[row * N2_ + col] = aR[v] * twi + aI[v] * twr;
    }
  }
  __syncthreads();

  // ---- stage 2: complex * complex, K = 128 ----
  for (int t = wave; t < 32; t += 8) {
    const int tm = t >> 3, tn = t & 7;
    const int k1 = tm * 16 + lm;
    const int k2 = tn * 16 + lm;
    v8f xR = {}, xI = {};
    for (int ks = 0; ks < 32; ++ks) {
      v2f azr, azi, ber, bei, bni;
#pragma unroll
      for (int j = 0; j < 2; ++j) {
        const int n2 = ks * 4 + lh * 2 + j;
        azr[j] = s_zr[k1 * N2_ + n2];
        azi[j] = s_zi[k1 * N2_ + n2];
        const int p = (n2 * k2) & 127;
        const float th = -TWOPI * (float)p * (1.0f / 128.0f);
        ber[j] = __cosf(th);
        bei[j] = __sinf(th);
        bni[j] = -bei[j];
      }
      xR = wmma4(azr, ber, xR);
      xR = wmma4(azi, bni, xR);  // - Zi*Ei
      xI = wmma4(azr, bei, xI);
      xI = wmma4(azi, ber, xI);
    }
#pragma unroll
    for (int v = 0; v < 8; ++v) {
      const int r1 = tm * 16 + v + lh * 8;   // k1
      const int c2 = tn * 16 + lm;           // k2
      const int k  = r1 + 64 * c2;
      if (k <= 4096) {
        Xr[(size_t)seq * FP_ + k] = xR[v];
        Xi[(size_t)seq * FP_ + k] = xI[v];
      }
    }
  }
}

// ---------------------------------------------------------------------------
// K3: logits -> gumbel -> softmax over freq, y_soft into ys, entropy to ent
// ---------------------------------------------------------------------------
__global__ __launch_bounds__(256) void logits_softmax(
    const float* __restrict__ Xr, const float* __restrict__ Xi,
    const float* __restrict__ diag, const float* __restrict__ bias,
    const float* __restrict__ un, float* __restrict__ ys,
    float* __restrict__ ent) {
  __shared__ float red[256];
  const int seq = blockIdx.x;
  const int bb  = seq / C_;
  const int cc  = seq % C_;
  const int tid = threadIdx.x;
  const float* d0  = diag;
  const float* d1  = diag + F_;
  const float* b0  = bias;
  const float* b1v = bias + F_;

  float mx = -3.0e38f;
  for (int k = tid; k < F_; k += 256) {
    const size_t o = (size_t)seq * FP_ + k;
    const float xr = Xr[o], xi = Xi[o];
    const float orr = fmaxf(xr * d0[k] - xi * d1[k] + b0[k], 0.0f);
    const float oii = fmaxf(xi * d0[k] + xr * d1[k] + b1v[k], 0.0f);
    const float lg  = sqrtf(orr * orr + oii * oii);
    const size_t uo = (size_t)bb * (size_t)F_ * C_ + (size_t)k * C_ + cc;
    __builtin_prefetch(&un[uo + 256u * C_], 0, 1);
    const float g = -__logf(-__logf(un[uo] + 1e-10f) + 1e-10f);
    const float v = (lg + g) * (1.0f / TAU_);
    ys[o] = v;
    mx = fmaxf(mx, v);
  }
  red[tid] = mx; __syncthreads();
  for (int s = 128; s > 0; s >>= 1) {
    if (tid < s) red[tid] = fmaxf(red[tid], red[tid + s]);
    __syncthreads();
  }
  mx = red[0]; __syncthreads();

  float sum = 0.0f;
  for (int k = tid; k < F_; k += 256) {
    const size_t o = (size_t)seq * FP_ + k;
    const float e = __expf(ys[o] - mx);
    ys[o] = e; sum += e;
  }
  red[tid] = sum; __syncthreads();
  for (int s = 128; s > 0; s >>= 1) {
    if (tid < s) red[tid] += red[tid + s];
    __syncthreads();
  }
  const float inv = 1.0f / red[0]; __syncthreads();

  float h = 0.0f;
  for (int k = tid; k < F_; k += 256) {
    const size_t o = (size_t)seq * FP_ + k;
    const float y = ys[o] * inv;
    ys[o] = y;
    h -= y * __logf(y);
  }
  red[tid] = h; __syncthreads();
  for (int s = 128; s > 0; s >>= 1) {
    if (tid < s) red[tid] += red[tid + s];
    __syncthreads();
  }
  if (tid == 0) atomicAdd(ent + bb, red[0] * (1.0f / C_));
}

// ---------------------------------------------------------------------------
// K4: per-(b,c) top-KTOP radix select on y_soft (positive floats: uint order),
//     then apply hard mask to spectrum in place (forward value of ST mask).
// ---------------------------------------------------------------------------
__global__ __launch_bounds__(256) void topk_mask(float* __restrict__ Xr,
                                                 float* __restrict__ Xi,
                                                 const float* __restrict__ ys) {
  __shared__ unsigned hist[256];
  __shared__ unsigned s_pref;
  __shared__ int s_rem;
  __shared__ int s_tie;
  const int seq = blockIdx.x;
  const int tid = threadIdx.x;
  if (tid == 0) { s_pref = 0u; s_rem = KTOP; }
  __syncthreads();

  for (int pass = 0; pass < 4; ++pass) {
    const int sh = 24 - pass * 8;
    hist[tid] = 0u;
    __syncthreads();
    const unsigned pref  = s_pref;
    const unsigned pmask = (pass == 0) ? 0u : (0xFFFFFFFFu << (sh + 8));
    for (int k = tid; k < F_; k += 256) {
      const unsigned u = __float_as_uint(ys[(size_t)seq * FP_ + k]);
      if ((u & pmask) == pref)
        atomicAdd(&hist[(u >> sh) & 255u], 1u);
    }
    __syncthreads();
    if (tid == 0) {
      int rem = s_rem;
      int bin = 255;
      for (; bin >= 0; --bin) {
        const int cnt = (int)hist[bin];
        if (rem <= cnt) break;
        rem -= cnt;
      }
      if (bin < 0) bin = 0;
      s_pref = s_pref | ((unsigned)bin << sh);
      s_rem  = rem;
    }
    __syncthreads();
  }
  const unsigned T = s_pref;
  const int ties   = s_rem;
  if (tid == 0) s_tie = 0;
  __syncthreads();

  for (int k = tid; k < F_; k += 256) {
    const size_t o = (size_t)seq * FP_ + k;
    const unsigned u = __float_as_uint(ys[o]);
    float m = 0.0f;
    if (u > T) m = 1.0f;
    else if (u == T && atomicAdd(&s_tie, 1) < ties) m = 1.0f;
    Xr[o] *= m;
    Xi[o] *= m;
  }
}

// ---------------------------------------------------------------------------
// K5: inverse four-step (Hermitian extension in LDS), f32 WMMA, 1/L scale.
//   stageA: G[k1][n2] = sum_k2 X[k1+64k2] e^{+2pi i k2 n2/128}
//   twiddle: G' = G * e^{+2pi i k1 n2/8192}
//   stageB: x[128 n1 + n2] = (1/L) Re( sum_k1 e^{+2pi i n1 k1/64} G'[k1][n2] )
// Masked spectrum rows staged into LDS by the TDM, mirrored LDS->LDS.
// ---------------------------------------------------------------------------
__global__ __launch_bounds__(256) void dft_inverse(const float* __restrict__ Xr,
                                                   const float* __restrict__ Xi,
                                                   float* __restrict__ xinv) {
  __shared__ float s_fr[L_];
  __shared__ float s_fi[L_];
  __shared__ float s_gr[N1_ * N2_];
  __shared__ float s_gi[N1_ * N2_];
  const int seq  = blockIdx.x;
  const int tid  = threadIdx.x;
  const int wave = tid >> 5;
  const int lane = tid & 31;
  const int lm   = lane & 15;
  const int lh   = lane >> 4;

#if HAVE_TDM
  if (wave == 0) {
    tdm_load_1d(Xr + (size_t)seq * FP_, s_fr, (unsigned)F_);
    tdm_load_1d(Xi + (size_t)seq * FP_, s_fi, (unsigned)F_);
    __builtin_amdgcn_s_wait_tensorcnt(0);
  }
  __syncthreads();
  for (int k = tid + 1; k < 4096; k += 256) {   // Hermitian mirror, LDS->LDS
    s_fr[L_ - k] = s_fr[k];
    s_fi[L_ - k] = -s_fi[k];
  }
#else
  for (int k = tid; k <= 4096; k += 256) {
    const float r = Xr[(size_t)seq * FP_ + k];
    const float i = Xi[(size_t)seq * FP_ + k];
    s_fr[k] = r; s_fi[k] = i;
    if (k >= 1 && k < 4096) { s_fr[L_ - k] = r; s_fi[L_ - k] = -i; }
  }
#endif
  __syncthreads();

  // ---- stage A: complex*complex, K = 128 ----
  for (int t = wave; t < 32; t += 8) {
    const int tm = t >> 3, tn = t & 7;
    const int k1  = tm * 16 + lm;
    const int n2c = tn * 16 + lm;
    v8f gR = {}, gI = {};
    for (int ks = 0; ks < 32; ++ks) {
      v2f axr, axi, ber, bei, bni;
#pragma unroll
      for (int j = 0; j < 2; ++j) {
        const int k2 = ks * 4 + lh * 2 + j;
        axr[j] = s_fr[k1 + 64 * k2];
        axi[j] = s_fi[k1 + 64 * k2];
        const int p = (k2 * n2c) & 127;
        const float th = TWOPI * (float)p * (1.0f / 128.0f);
        ber[j] = __cosf(th);
        bei[j] = __sinf(th);
        bni[j] = -bei[j];
      }
      gR = wmma4(axr, ber, gR);
      gR = wmma4(axi, bni, gR);
      gI = wmma4(axr, bei, gI);
      gI = wmma4(axi, ber, gI);
    }
#pragma unroll
    for (int v = 0; v < 8; ++v) {
      const int row = tm * 16 + v + lh * 8;   // k1
      const int col = tn * 16 + lm;           // n2
      const int p   = (row * col) & 8191;
      const float th  = TWOPI * (float)p * (1.0f / 8192.0f);
      const float twr = __cosf(th), twi = __sinf(th);
      s_gr[row * N2_ + col] = gR[v] * twr - gI[v] * twi;
      s_gi[row * N2_ + col] = gR[v] * twi + gI[v] * twr;
    }
  }
  __syncthreads();

  // ---- stage B: real output, K = 64 ----
  for (int t = wave; t < 32; t += 8) {
    const int tm = t >> 3, tn = t & 7;
    const int n1  = tm * 16 + lm;
    const int n2c = tn * 16 + lm;
    v8f xo = {};
    for (int ks = 0; ks < 16; ++ks) {
      v2f ac, ans, bgr, bgi;
#pragma unroll
      for (int j = 0; j < 2; ++j) {
        const int k1 = ks * 4 + lh * 2 + j;
        const int p  = (n1 * k1) & 63;
        const float th = TWOPI * (float)p * (1.0f / 64.0f);
        ac[j]  = __cosf(th);
        ans[j] = -__sinf(th);
        bgr[j] = s_gr[k1 * N2_ + n2c];
        bgi[j] = s_gi[k1 * N2_ + n2c];
      }
      xo = wmma4(ac, bgr, xo);   // cos*Gr
      xo = wmma4(ans, bgi, xo);  // -sin*Gi
    }
#pragma unroll
    for (int v = 0; v < 8; ++v) {
      const int r = tm * 16 + v + lh * 8;     // n1
      const int c = tn * 16 + lm;             // n2
      xinv[(size_t)seq * L_ + r * N2_ + c] = xo[v] * (1.0f / (float)L_);
    }
  }
}

// ---------------------------------------------------------------------------
// K6: emit x_var[b,l,c] = x - x_inv and x_inv[b,l,c]  (tiled transpose back)
// ---------------------------------------------------------------------------
__global__ __launch_bounds__(256) void finalize(const float* __restrict__ x,
                                                const float* __restrict__ xinv,
                                                float* __restrict__ x_var,
                                                float* __restrict__ x_inv_out) {
  __shared__ float tile[32][33];
  const int l0 = blockIdx.x * 32;
  const int b  = blockIdx.y;
  const int r  = threadIdx.x >> 5;
  const int c  = threadIdx.x & 31;
  for (int i = r; i < 32; i += 8)           // i = channel, c = l offset
    tile[i][c] = xinv[((size_t)(b * C_ + i)) * L_ + l0 + c];
  __syncthreads();
  for (int i = r; i < 32; i += 8) {         // i = l offset, c = channel
    const size_t o = ((size_t)b * L_ + l0 + i) * C_ + c;
    const float inv = tile[c][i];
    x_var[o]     = x[o] - inv;
    x_inv_out[o] = inv;
  }
}

// ---------------------------------------------------------------------------
extern "C" void kernel_launch(void* const* d_in, const int* in_sizes, int n_in,
                              void* d_out, int out_size, void* d_ws,
                              size_t ws_size, hipStream_t stream) {
  (void)in_sizes; (void)n_in; (void)out_size; (void)ws_size;
  const float* x  = (const float*)d_in[0];   // [B,L,C]
  const float* w1 = (const float*)d_in[1];   // [2,F,F]
  const float* b1 = (const float*)d_in[2];   // [2,F]
  const float* un = (const float*)d_in[3];   // [B,F,C]

  float* x_var = (float*)d_out;
  float* x_inv = x_var + (size_t)B_ * L_ * C_;
  float* ent   = x_inv + (size_t)B_ * L_ * C_;

  float* wsf  = (float*)d_ws;
  float* xT   = wsf;                               // S*L, reused as xinv_seq
  float* Xr   = xT + (size_t)S_ * L_;              // S*FP
  float* Xi   = Xr + (size_t)S_ * FP_;             // S*FP
  float* ys   = Xi + (size_t)S_ * FP_;             // S*FP
  float* diag = ys + (size_t)S_ * FP_;             // 2*F

  const dim3 tgrid(L_ / 32, B_);
  hipLaunchKernelGGL(transpose_in, tgrid, dim3(256), 0, stream, x, xT);
  hipLaunchKernelGGL(diag_extract, dim3((2 * F_ + 255) / 256), dim3(256), 0,
                     stream, w1, diag);
  hipLaunchKernelGGL(dft_forward, dim3(S_), dim3(256), 0, stream, xT, Xr, Xi);
  hipLaunchKernelGGL(ent_init, dim3(1), dim3(64), 0, stream, ent);
  hipLaunchKernelGGL(logits_softmax, dim3(S_), dim3(256), 0, stream, Xr, Xi,
                     diag, b1, un, ys, ent);
  hipLaunchKernelGGL(topk_mask, dim3(S_), dim3(256), 0, stream, Xr, Xi, ys);
  hipLaunchKernelGGL(dft_inverse, dim3(S_), dim3(256), 0, stream, Xr, Xi, xT);
  hipLaunchKernelGGL(finalize, tgrid, dim3(256), 0, stream, x, xT, x_var,
                     x_inv);
}